// LayerNormSoftmaxChain_2757369004393
// MI455X (gfx1250) — compile-verified
//
#include <hip/hip_runtime.h>
#include <hip/hip_bf16.h>

typedef __attribute__((ext_vector_type(2))) float v2f;
typedef __attribute__((ext_vector_type(4))) float v4f;
typedef __attribute__((ext_vector_type(8))) float v8f;

#define LN_EPS 1e-5f
#define LOG2E  1.44269504088896340736f

// LayerNorm(4) -> Linear(4->3) via V_WMMA_F32_16X16X4_F32 -> softmax(3).
// One row per lane; wave handles 32 rows with 2 WMMAs.
//
// A matrix (16x4 f32, per doc layout: lanes0-15 M=0..15; VGPR0 = K=0|K=2,
// VGPR1 = K=1|K=3 across lane halves): A[m][k] = W[k][m%8] for
// m in {0,1,2, 8,9,10}, else 0.  The M=8..10 duplicate makes lanes 16-31
// of D (which read M=j+8) produce valid logits too.
//
// B matrix (4x16 f32, 2 VGPRs, same half-lane K split): column n = one row
// of normalized x. Built from per-lane h[0..3] with 4 lane^16 swizzles.
//
// D: VGPR j, lanes0-15 -> M=j, N=lane; lanes16-31 -> M=j+8, N=lane-16.
// So logit_j(row lane) = D1[j] for lanes<16 and D2[j] for lanes>=16.
__global__ void __launch_bounds__(256)
lnsm_wmma_kernel(const float* __restrict__ x,
                 const float* __restrict__ W,
                 const float* __restrict__ gamma,
                 const float* __restrict__ beta,
                 float* __restrict__ out, int n) {
  const int tid  = blockIdx.x * blockDim.x + threadIdx.x;
  const int lane = threadIdx.x & 31;
  const bool lo  = lane < 16;
  const int row  = (tid < n) ? tid : (n - 1);   // clamp: keep EXEC all-1s for WMMA

  // ---- uniform params (scalar loads, L0-const cached) ----
  const float g0 = gamma[0], g1 = gamma[1], g2 = gamma[2], g3 = gamma[3];
  const float b0 = beta[0],  b1 = beta[1],  b2 = beta[2],  b3 = beta[3];

  // ---- stream one row: global_load_b128, non-temporal ----
  v4f xv = __builtin_nontemporal_load((const v4f*)x + row);

  // ---- LayerNorm over the 4 elements, fully in-lane ----
  const float mu = (xv.x + xv.y + xv.z + xv.w) * 0.25f;
  const float d0 = xv.x - mu, d1 = xv.y - mu, d2 = xv.z - mu, d3 = xv.w - mu;
  const float var = (d0 * d0 + d1 * d1 + d2 * d2 + d3 * d3) * 0.25f;
  const float inv = __builtin_amdgcn_rsqf(var + LN_EPS);
  const float h0 = d0 * inv * g0 + b0;
  const float h1 = d1 * inv * g1 + b1;
  const float h2 = d2 * inv * g2 + b2;
  const float h3 = d3 * inv * g3 + b3;

  // ---- A matrix from W (per-lane, loaded once; W is 4x3 row-major) ----
  const int mA   = lane & 15;
  const int kA   = lo ? 0 : 2;
  const int col  = mA & 7;
  const bool okA = col < 3;                    // m in {0,1,2,8,9,10}
  v2f A;
  A.x = okA ? W[kA * 3 + col]       : 0.0f;
  A.y = okA ? W[(kA + 1) * 3 + col] : 0.0f;

  // ---- build B matrices: 4 x SWAPX16 lane^16 swizzles + selects ----
  // ds_swizzle imm 0x401F: group-of-32, and=0x1f, or=0, xor=0x10 -> lane^16
  const float h0s = __int_as_float(__builtin_amdgcn_ds_swizzle(__float_as_int(h0), 0x401F));
  const float h1s = __int_as_float(__builtin_amdgcn_ds_swizzle(__float_as_int(h1), 0x401F));
  const float h2s = __int_as_float(__builtin_amdgcn_ds_swizzle(__float_as_int(h2), 0x401F));
  const float h3s = __int_as_float(__builtin_amdgcn_ds_swizzle(__float_as_int(h3), 0x401F));

  v2f B1;                  // columns = rows base+0 .. base+15
  B1.x = lo ? h0 : h2s;    // K=0 | K=2
  B1.y = lo ? h1 : h3s;    // K=1 | K=3
  v2f B2;                  // columns = rows base+16 .. base+31
  B2.x = lo ? h0s : h2;
  B2.y = lo ? h1s : h3;

  // ---- D = A x B + 0 : two V_WMMA_F32_16X16X4_F32 per wave ----
  v8f c = {};
  v8f D1 = __builtin_amdgcn_wmma_f32_16x16x4_f32(false, A, false, B1,
                                                 (short)0, c, false, false);
  v8f D2 = __builtin_amdgcn_wmma_f32_16x16x4_f32(false, A, false, B2,
                                                 (short)0, c, false, false);

  // ---- logits land in-lane: lanes<16 from D1 (M=0..2), lanes>=16 from D2 (M=8..10) ----
  const float l0 = lo ? D1[0] : D2[0];
  const float l1 = lo ? D1[1] : D2[1];
  const float l2 = lo ? D1[2] : D2[2];

  // ---- softmax over 3, per-lane ----
  const float mx = fmaxf(l0, fmaxf(l1, l2));
  const float e0 = __builtin_amdgcn_exp2f((l0 - mx) * LOG2E);
  const float e1 = __builtin_amdgcn_exp2f((l1 - mx) * LOG2E);
  const float e2 = __builtin_amdgcn_exp2f((l2 - mx) * LOG2E);
  const float rs = __builtin_amdgcn_rcpf(e0 + e1 + e2);

  if (tid < n) {
    float* o = out + (size_t)tid * 3;
    __builtin_nontemporal_store(e0 * rs, o + 0);
    __builtin_nontemporal_store(e1 * rs, o + 1);
    __builtin_nontemporal_store(e2 * rs, o + 2);
  }
}

extern "C" void kernel_launch(void* const* d_in, const int* in_sizes, int n_in,
                              void* d_out, int out_size, void* d_ws, size_t ws_size,
                              hipStream_t stream) {
  const float* x     = (const float*)d_in[0];   // [N,4]
  const float* W     = (const float*)d_in[1];   // [4,3]
  const float* gamma = (const float*)d_in[2];   // [4]
  const float* beta  = (const float*)d_in[3];   // [4]
  float* out         = (float*)d_out;           // [N,3]

  const int n = in_sizes[0] / 4;                // number of rows
  if (n <= 0) return;

  const int block = 256;
  const int grid  = (n + block - 1) / block;
  lnsm_wmma_kernel<<<grid, block, 0, stream>>>(x, W, gamma, beta, out, n);
}